// KMeans_7198365188303
// MI455X (gfx1250) — compile-verified
//
#include <hip/hip_runtime.h>
#include <stdint.h>

// Problem constants (match reference)
#define NN 131072
#define DD 128
#define KK 512

#define TILE_N   128   // input columns per workgroup
#define CHUNK_K  128   // centroid rows staged in LDS per iteration
#define LDS_STRIDE 132 // 128 + 4 pad -> stride mod 64 banks == 4: conflict-free b64 frags

typedef float v2f __attribute__((ext_vector_type(2)));
typedef float v4f __attribute__((ext_vector_type(4)));
typedef float v8f __attribute__((ext_vector_type(8)));

static constexpr int    kCsFloats = CHUNK_K * LDS_STRIDE;
static constexpr int    kXsFloats = TILE_N * LDS_STRIDE;
static constexpr size_t kSmemBytes =
    sizeof(unsigned long long) * TILE_N                          // argmin keys
  + sizeof(float) * (size_t)(kCsFloats + kXsFloats + CHUNK_K + TILE_N);
static_assert(kSmemBytes <= 320u * 1024u, "fits WGP LDS");

__device__ __forceinline__ unsigned int order_f32(float f) {
  // order-preserving float -> uint (handles negatives from rounding)
  unsigned int u = __float_as_uint(f);
  return (u & 0x80000000u) ? ~u : (u | 0x80000000u);
}

__global__ __launch_bounds__(256) void kmeans_wmma_f32(
    const float* __restrict__ inputs,     // [N, D]
    const float* __restrict__ centroids,  // [K, D]
    float* __restrict__ dist_out,         // [K, N]
    float* __restrict__ assign_out) {     // [N] (float-encoded indices)
  extern __shared__ unsigned char smem_raw[];
  unsigned long long* best = (unsigned long long*)smem_raw;
  float* Cs = (float*)(best + TILE_N);   // centroid chunk [CHUNK_K][LDS_STRIDE]
  float* Xs = Cs + kCsFloats;            // input tile     [TILE_N][LDS_STRIDE]
  float* cq = Xs + kXsFloats;            // ||c||^2 for chunk
  float* xq = cq + CHUNK_K;              // ||x||^2 for tile

  const int tid   = threadIdx.x;
  const int lane  = tid & 31;
  const int wave  = tid >> 5;
  const int half  = lane >> 4;   // 0: lanes 0-15, 1: lanes 16-31
  const int r     = lane & 15;
  const int kw    = wave >> 1;   // 4 k-waves  -> 32 k each
  const int nw    = wave & 1;    // 2 n-waves  -> 64 n each
  const int nbase = blockIdx.x * TILE_N;

  // ---- stage input tile (streamed once; non-temporal so L2 keeps centroids) ----
  for (int i = tid; i < TILE_N * (DD / 4); i += 256) {
    const int row = i >> 5, c4 = (i & 31) * 4;
    v4f v = __builtin_nontemporal_load(
        (const v4f*)(inputs + (size_t)(nbase + row) * DD + c4));
    *(v4f*)(Xs + row * LDS_STRIDE + c4) = v;
  }
  if (tid < TILE_N) best[tid] = ~0ull;
  __syncthreads();
  if (tid < TILE_N) {
    const float* xrow = Xs + tid * LDS_STRIDE;
    float s = 0.f;
    #pragma unroll 8
    for (int d = 0; d < DD; ++d) s += xrow[d] * xrow[d];
    xq[tid] = s;
  }

  for (int kk = 0; kk < KK / CHUNK_K; ++kk) {
    // ---- stage centroid chunk (L2-resident broadcast) ----
    for (int i = tid; i < CHUNK_K * (DD / 4); i += 256) {
      const int row = i >> 5, c4 = (i & 31) * 4;
      v4f v = *(const v4f*)(centroids + (size_t)(kk * CHUNK_K + row) * DD + c4);
      *(v4f*)(Cs + row * LDS_STRIDE + c4) = v;
    }
    // prefetch next chunk while this one is consumed (global_prefetch_b8)
    if (kk + 1 < KK / CHUNK_K) {
      const char* nxt = (const char*)(centroids + (size_t)(kk + 1) * CHUNK_K * DD);
      for (int off = tid * 128; off < CHUNK_K * DD * 4; off += 256 * 128)
        __builtin_prefetch(nxt + off, 0, 3);
    }
    __syncthreads();
    if (tid < CHUNK_K) {
      const float* crow = Cs + tid * LDS_STRIDE;
      float s = 0.f;
      #pragma unroll 8
      for (int d = 0; d < DD; ++d) s += crow[d] * crow[d];
      cq[tid] = s;
    }
    __syncthreads();

    // ---- 32k x 64n register tile per wave: 2x4 WMMA f32 accumulators ----
    v8f acc[2][4];
    #pragma unroll
    for (int mt = 0; mt < 2; ++mt)
      #pragma unroll
      for (int nt = 0; nt < 4; ++nt)
        acc[mt][nt] = (v8f){0.f, 0.f, 0.f, 0.f, 0.f, 0.f, 0.f, 0.f};

    const float* Abase = Cs + (kw * 32 + r) * LDS_STRIDE + 2 * half; // A: lanes 0-15 K=d..d+1, 16-31 K=d+2..d+3
    const float* Bbase = Xs + (nw * 64 + r) * LDS_STRIDE + 2 * half; // B: same K split, N=lane&15

    #pragma unroll 8
    for (int d = 0; d < DD; d += 4) {
      v2f a0 = *(const v2f*)(Abase + 0 * 16 * LDS_STRIDE + d);
      v2f a1 = *(const v2f*)(Abase + 1 * 16 * LDS_STRIDE + d);
      v2f b0 = *(const v2f*)(Bbase + 0 * 16 * LDS_STRIDE + d);
      v2f b1 = *(const v2f*)(Bbase + 1 * 16 * LDS_STRIDE + d);
      v2f b2 = *(const v2f*)(Bbase + 2 * 16 * LDS_STRIDE + d);
      v2f b3 = *(const v2f*)(Bbase + 3 * 16 * LDS_STRIDE + d);
      acc[0][0] = __builtin_amdgcn_wmma_f32_16x16x4_f32(false, a0, false, b0, (short)0, acc[0][0], false, false);
      acc[0][1] = __builtin_amdgcn_wmma_f32_16x16x4_f32(false, a0, false, b1, (short)0, acc[0][1], false, false);
      acc[0][2] = __builtin_amdgcn_wmma_f32_16x16x4_f32(false, a0, false, b2, (short)0, acc[0][2], false, false);
      acc[0][3] = __builtin_amdgcn_wmma_f32_16x16x4_f32(false, a0, false, b3, (short)0, acc[0][3], false, false);
      acc[1][0] = __builtin_amdgcn_wmma_f32_16x16x4_f32(false, a1, false, b0, (short)0, acc[1][0], false, false);
      acc[1][1] = __builtin_amdgcn_wmma_f32_16x16x4_f32(false, a1, false, b1, (short)0, acc[1][1], false, false);
      acc[1][2] = __builtin_amdgcn_wmma_f32_16x16x4_f32(false, a1, false, b2, (short)0, acc[1][2], false, false);
      acc[1][3] = __builtin_amdgcn_wmma_f32_16x16x4_f32(false, a1, false, b3, (short)0, acc[1][3], false, false);
    }

    // ---- epilogue: dist = ||c||^2 + ||x||^2 - 2*cross ; NT stream out ; argmin ----
    #pragma unroll
    for (int mt = 0; mt < 2; ++mt) {
      #pragma unroll
      for (int nt = 0; nt < 4; ++nt) {
        const int    ncol  = nw * 64 + nt * 16 + r;
        const size_t nglob = (size_t)nbase + ncol;
        const float  xsv   = xq[ncol];
        unsigned long long bk = ~0ull;
        #pragma unroll
        for (int v = 0; v < 8; ++v) {
          const int klocal = kw * 32 + mt * 16 + half * 8 + v; // D-layout: M = v + 8*half
          const int kglob  = kk * CHUNK_K + klocal;
          const float dist = cq[klocal] + xsv - 2.0f * acc[mt][nt][v];
          __builtin_nontemporal_store(dist, dist_out + (size_t)kglob * NN + nglob);
          const unsigned long long key =
              ((unsigned long long)order_f32(dist) << 32) | (unsigned int)kglob;
          bk = key < bk ? key : bk;
        }
        atomicMin(best + ncol, bk); // ds_min_u64; tie -> smallest k (matches jnp.argmin)
      }
    }
    __syncthreads(); // protect Cs/cq restage and final best readout
  }

  if (tid < TILE_N) {
    const unsigned int kbest = (unsigned int)(best[tid] & 0xFFFFFFFFull);
    assign_out[nbase + tid] = (float)kbest;
  }
}

extern "C" void kernel_launch(void* const* d_in, const int* in_sizes, int n_in,
                              void* d_out, int out_size, void* d_ws, size_t ws_size,
                              hipStream_t stream) {
  (void)in_sizes; (void)n_in; (void)out_size; (void)d_ws; (void)ws_size;
  const float* inputs    = (const float*)d_in[0];
  const float* centroids = (const float*)d_in[1];
  float* dist_out   = (float*)d_out;
  float* assign_out = dist_out + (size_t)KK * NN;

  (void)hipFuncSetAttribute((const void*)kmeans_wmma_f32,
                            hipFuncAttributeMaxDynamicSharedMemorySize,
                            (int)kSmemBytes);
  dim3 grid(NN / TILE_N);   // 1024 workgroups
  kmeans_wmma_f32<<<grid, 256, kSmemBytes, stream>>>(inputs, centroids,
                                                     dist_out, assign_out);
}